// Ordered_GCN_45286135169449
// MI455X (gfx1250) — compile-verified
//
#include <hip/hip_runtime.h>

typedef __attribute__((ext_vector_type(2))) float v2f;
typedef __attribute__((ext_vector_type(4))) float v4f;
typedef __attribute__((ext_vector_type(8))) float v8f;

#define B_    64
#define N_    207
#define K_    32
#define C_    8
#define DIN   64
#define DOUT  64
#define ROWS  16                    // (b,n) rows per workgroup
#define RSTRIDE (C_ * DIN + 4)      // 516 floats: pad -> 4-bank stagger per row
#define OSTRIDE (C_ * DOUT + 4)     // 516 floats: same stagger for output staging

// Hardware tanh (gfx1250 TRANS op) when the builtin exists; precise fallback.
#if __has_builtin(__builtin_amdgcn_tanhf)
  #define TANH(x) __builtin_amdgcn_tanhf(x)
#elif __has_builtin(__builtin_amdgcn_tanh_f32)
  #define TANH(x) __builtin_amdgcn_tanh_f32(x)
#else
  #define TANH(x) tanhf(x)
#endif

__global__ __launch_bounds__(256)
void ordered_gcn_kernel(const int*   __restrict__ idx,   // [B,N,K]
                        const float* __restrict__ wd,    // [B,N,K,DIN]
                        const float* __restrict__ W,     // [C,DOUT,DIN]
                        float*       __restrict__ out)   // [B,N,C,DOUT]
{
    __shared__ float s_mean[ROWS * RSTRIDE];   // ~32.3 KB
    __shared__ float s_out [ROWS * OSTRIDE];   // ~32.3 KB output staging
    __shared__ float s_rcnt[ROWS][C_];
    __shared__ int   s_idx[ROWS][K_];

    const int  tid  = threadIdx.x;
    const long row0 = (long)blockIdx.x * ROWS;   // global (b*n) row base

    // ---- load indices for the 16 rows (512 ints) + zero the mean buffer ----
    for (int i = tid; i < ROWS * K_; i += 256)
        s_idx[i / K_][i % K_] = idx[row0 * K_ + i];
    for (int i = tid; i < ROWS * RSTRIDE; i += 256)
        s_mean[i] = 0.0f;
    __syncthreads();

    // ---- per (row, class) reciprocal clamped counts (128 threads) ----
    if (tid < ROWS * C_) {
        const int r = tid / C_, c = tid % C_;
        int cnt = 0;
        #pragma unroll
        for (int k = 0; k < K_; ++k) cnt += (s_idx[r][k] == c);
        s_rcnt[r][c] = 1.0f / (float)(cnt < 1 ? 1 : cnt);
    }

    // ---- scatter-sum: 16 lanes per row own 4 channels each (conflict-free) ----
    {
        const int r  = tid >> 4;           // 0..15
        const int dq = (tid & 15) << 2;    // 0,4,...,60
        const float* src = wd + (row0 + r) * (long)(K_ * DIN) + dq;
        float* mrow = s_mean + r * RSTRIDE;
        #pragma unroll 8
        for (int k = 0; k < K_; ++k) {
            const v4f v = *(const v4f*)(src + k * DIN);
            const int c = s_idx[r][k];
            float* m = mrow + c * DIN + dq;
            m[0] += v.x; m[1] += v.y; m[2] += v.z; m[3] += v.w;
        }
    }
    __syncthreads();

    // ---- sums -> means ----
    for (int i = tid; i < ROWS * C_ * DIN; i += 256) {
        const int r   = i >> 9;        // /512
        const int rem = i & 511;
        const int c   = rem >> 6;      // /64
        s_mean[r * RSTRIDE + rem] *= s_rcnt[r][c];
    }
    __syncthreads();

    // ---- phase 2: per-class GEMM via chained fp32 WMMA, then hw tanh ----
    // 32 output tiles (8 classes x 4 N-tiles of 16); 8 waves -> 4 tiles each.
    const int wave = tid >> 5;
    const int lane = tid & 31;
    const int lm   = lane & 15;        // A: M row / B,D: N column
    const int lh   = lane >> 4;        // half-wave select (K sub-pair)

    for (int t = wave; t < C_ * 4; t += 8) {
        const int c  = t >> 2;
        const int nb = (t & 3) * 16;   // N-tile base within DOUT

        v8f acc = {};
        // this lane's B column: W[c][nb+lm][*] (row of W, stride DIN)
        const float* wcol = W + ((long)c * DOUT + (nb + lm)) * DIN;
        const float* arow = s_mean + lm * RSTRIDE + c * DIN;

        #pragma unroll
        for (int kk = 0; kk < 16; ++kk) {
            const int kb = kk * 4;
            // A 16x4 fp32 layout: lanes 0-15 hold K+0,K+1; lanes 16-31 K+2,K+3
            v2f a = *(const v2f*)(arow + kb + lh * 2);
            // B 4x16 fp32 layout mirrors A across lane halves
            v2f b = *(const v2f*)(wcol + kb + lh * 2);
            acc = __builtin_amdgcn_wmma_f32_16x16x4_f32(
                      false, a, false, b, (short)0, acc, false, false);
        }

        // D layout: VGPR j -> M = j (lanes 0-15) / j+8 (lanes 16-31); N = lm
        // Stage tanh(D) into LDS for coalesced streaming stores.
        #pragma unroll
        for (int j = 0; j < 8; ++j) {
            const int m = j + lh * 8;
            s_out[m * OSTRIDE + c * DOUT + nb + lm] = TANH(acc[j]);
        }
    }
    __syncthreads();

    // ---- coalesced b128 store: block output is one contiguous 32 KB span ----
    {
        float* ob = out + row0 * (C_ * DOUT);
        #pragma unroll
        for (int i = tid * 4; i < ROWS * C_ * DOUT; i += 256 * 4) {
            const int r   = i >> 9;        // /512
            const int off = i & 511;
            *(v4f*)(ob + i) = *(const v4f*)(s_out + r * OSTRIDE + off);
        }
    }
}

extern "C" void kernel_launch(void* const* d_in, const int* in_sizes, int n_in,
                              void* d_out, int out_size, void* d_ws, size_t ws_size,
                              hipStream_t stream) {
    const int*   idx = (const int*)  d_in[0];   // clustered_index_topk [B,N,K] int32
    const float* wd  = (const float*)d_in[1];   // weightedDinput_topk  [B,N,K,DIN]
    const float* W   = (const float*)d_in[2];   // W                    [C,DOUT,DIN]
    float*       out = (float*)      d_out;     // [B,N,C,DOUT]

    const int rows  = B_ * N_;                  // 13248
    const int grid  = (rows + ROWS - 1) / ROWS; // 828
    ordered_gcn_kernel<<<grid, 256, 0, stream>>>(idx, wd, W, out);
}